// Sparsify1D_kactiveIOnline_51848845197802
// MI455X (gfx1250) — compile-verified
//
#include <hip/hip_runtime.h>
#include <stdint.h>

// Per-row top-K thresholding:  out = x * (x >= kth_largest_per_row(x))
// rows = 16384, cols = 4096, K = int(4096 * 0.8) = 3276
#define COLS   4096
#define KKEEP  3276
#define TPB    256      // 8 waves (wave32)
#define V4PT   4        // float4 chunks per thread  (4 * 256 * 4 = 4096 floats)
#define EPT    16       // elements per thread
#define NWAVE  8

__device__ __forceinline__ uint32_t key_of(float f) {
    // Order-preserving monotone transform: bigger float <=> bigger uint key.
    uint32_t u = __float_as_uint(f);
    return (u & 0x80000000u) ? ~u : (u | 0x80000000u);
}

__global__ __launch_bounds__(TPB)
void sparsify_topk_kernel(const float* __restrict__ x,
                          float* __restrict__ out,
                          int rows) {
    __shared__ __align__(16) float rowbuf[COLS];      // 16 KB staged row
    __shared__ uint32_t hist[NWAVE][256];             // 8 KB per-wave histograms
    __shared__ uint32_t suf[256];                     // suffix-scan workspace
    __shared__ uint32_t bcastDigit;
    __shared__ uint32_t bcastRem;

    const int t    = threadIdx.x;
    const int wave = t >> 5;
    const long long rowBase = (long long)blockIdx.x * COLS;
    const float4* __restrict__ gin  = (const float4*)(x + rowBase);
    float4* __restrict__       gout = (float4*)(out + rowBase);

    // ---- Stage the row into LDS with CDNA5 async global->LDS copies ----
    #pragma unroll
    for (int c = 0; c < V4PT; ++c) {
        const int v4 = t + c * TPB;                               // 0..1023
        uint32_t loff = (uint32_t)(uintptr_t)(&rowbuf[v4 * 4]);   // LDS byte addr
        const void* g = (const void*)(gin + v4);
        asm volatile("global_load_async_to_lds_b128 %0, %1, off"
                     :: "v"(loff), "v"(g) : "memory");
    }
    asm volatile("s_wait_asynccnt 0" ::: "memory");
    __syncthreads();

    // ---- Pull my 16 elements into registers, build sortable keys ----
    float    vals[EPT];
    uint32_t keys[EPT];
    #pragma unroll
    for (int c = 0; c < V4PT; ++c) {
        float4 q = ((const float4*)rowbuf)[t + c * TPB];
        vals[c * 4 + 0] = q.x; vals[c * 4 + 1] = q.y;
        vals[c * 4 + 2] = q.z; vals[c * 4 + 3] = q.w;
    }
    #pragma unroll
    for (int i = 0; i < EPT; ++i) keys[i] = key_of(vals[i]);

    // ---- 4-pass MSB-first radix select of the K-th largest key ----
    uint32_t prefix = 0, pmask = 0, remaining = KKEEP;
    #pragma unroll
    for (int pass = 0; pass < 4; ++pass) {
        const int shift = 24 - pass * 8;

        // zero per-wave histograms (each thread clears 8 entries)
        #pragma unroll
        for (int w = 0; w < NWAVE; ++w) hist[w][t] = 0;
        __syncthreads();

        // histogram of next digit, restricted to current prefix
        #pragma unroll
        for (int i = 0; i < EPT; ++i) {
            uint32_t k = keys[i];
            if ((k & pmask) == prefix)
                atomicAdd(&hist[wave][(k >> shift) & 0xFFu], 1u);
        }
        __syncthreads();

        // combine wave copies; thread t owns digit t
        uint32_t v = 0;
        #pragma unroll
        for (int w = 0; w < NWAVE; ++w) v += hist[w][t];
        suf[t] = v;
        __syncthreads();

        // inclusive suffix sum over 256 digits (Hillis-Steele, 8 steps)
        #pragma unroll
        for (int s = 1; s < 256; s <<= 1) {
            uint32_t add = (t + s < 256) ? suf[t + s] : 0u;
            __syncthreads();
            suf[t] += add;
            __syncthreads();
        }

        const uint32_t inc = suf[t];     // count of elems with digit >= t
        const uint32_t gt  = inc - v;    // count of elems with digit >  t
        if (v != 0u && gt < remaining && remaining <= inc) {
            bcastDigit = (uint32_t)t;    // exactly one thread matches
            bcastRem   = remaining - gt;
        }
        __syncthreads();

        prefix   |= bcastDigit << shift;
        pmask    |= 0xFFu << shift;
        remaining = bcastRem;
        __syncthreads();
    }

    const uint32_t tkey = prefix;        // exact key of the K-th largest

    // ---- Masked write-back (bit-exact with reference's mask * x) ----
    #pragma unroll
    for (int c = 0; c < V4PT; ++c) {
        float4 q;
        q.x = vals[c * 4 + 0] * ((keys[c * 4 + 0] >= tkey) ? 1.0f : 0.0f);
        q.y = vals[c * 4 + 1] * ((keys[c * 4 + 1] >= tkey) ? 1.0f : 0.0f);
        q.z = vals[c * 4 + 2] * ((keys[c * 4 + 2] >= tkey) ? 1.0f : 0.0f);
        q.w = vals[c * 4 + 3] * ((keys[c * 4 + 3] >= tkey) ? 1.0f : 0.0f);
        gout[t + c * TPB] = q;
    }
    (void)rows;
}

extern "C" void kernel_launch(void* const* d_in, const int* in_sizes, int n_in,
                              void* d_out, int out_size, void* d_ws, size_t ws_size,
                              hipStream_t stream) {
    const float* x = (const float*)d_in[0];
    float* out     = (float*)d_out;
    const int rows = in_sizes[0] / COLS;   // 16384
    dim3 grid(rows), block(TPB);
    hipLaunchKernelGGL(sparsify_topk_kernel, grid, block, 0, stream, x, out, rows);
    (void)n_in; (void)out_size; (void)d_ws; (void)ws_size;
}